// SkeletonEnergy_4398046511112
// MI455X (gfx1250) — compile-verified
//
#include <hip/hip_runtime.h>
#include <stdint.h>

// ---------------------------------------------------------------------------
// Soft skeletonization (clDice) fused per-iteration kernel for gfx1250.
//   pool(x) = (sum_{3x3} w*x)/(sum w),  w = exp(sign*x/alpha), alpha=0.05
//   sentinel -1e9 => w = 0 (matches reference NEG padding exactly).
// Tile 64x64, halo 3, staged to LDS with global_load_async_to_lds_b32
// (ASYNCcnt) — the CDNA5 async-tensor data path. wave32, 256 thr = 8 waves.
// ---------------------------------------------------------------------------

#define IMG_W   1024
#define IMG_H   1024
#define IMG_B   8
#define TILE    64
#define HALO    3
#define STILE   (TILE + 2 * HALO)   // 70
#define SP      72                  // padded LDS row stride (floats)
#define NEGS    (-1.0e9f)
#define VALID   (-1.0e8f)           // v > VALID  <=>  in-image value
#define INV_A   20.0f               // 1/alpha

__device__ __forceinline__ float wavg9(const float* __restrict__ v,
                                       const float* __restrict__ w, int idx) {
  float num = 0.f, den = 0.f;
#pragma unroll
  for (int dy = -1; dy <= 1; ++dy) {
#pragma unroll
    for (int dx = -1; dx <= 1; ++dx) {
      const int j = idx + dy * SP + dx;
      const float ww = w[j];
      num = fmaf(ww, v[j], num);
      den += ww;
    }
  }
  // caller guarantees center weight > 0  =>  den > 0
  return num / den;
}

template <bool INIT>
__global__ __launch_bounds__(256) void skel_iter(
    const float* __restrict__ img_in, const float* __restrict__ skel_in,
    float* __restrict__ img_out, float* __restrict__ skel_out) {
  __shared__ float s_a[STILE * SP];   // staged img, later t2
  __shared__ float s_b[STILE * SP];   // t1
  __shared__ float s_w[STILE * SP];   // softmax weights (exp plane)

  const int tid = threadIdx.x;
  const int tilesX = IMG_W / TILE;                    // 16
  const int bx = blockIdx.x % tilesX;
  const int by = (blockIdx.x / tilesX) % tilesX;
  const int b  = blockIdx.x / (tilesX * tilesX);
  const size_t base = (size_t)b * (IMG_W * IMG_H);
  const int x0 = bx * TILE - HALO;
  const int y0 = by * TILE - HALO;

  // ---- stage 70x70 tile (halo 3) into LDS via async global->LDS copies ----
  for (int i = tid; i < STILE * STILE; i += 256) {
    const int ly = i / STILE, lx = i - ly * STILE;
    const int gx = x0 + lx, gy = y0 + ly;
    const int didx = ly * SP + lx;
    if ((unsigned)gx < (unsigned)IMG_W && (unsigned)gy < (unsigned)IMG_H) {
      const unsigned loff =
          (unsigned)(uintptr_t)&s_a[didx];            // LDS byte offset
      const unsigned long long gaddr =
          (unsigned long long)(uintptr_t)(img_in + base + (size_t)gy * IMG_W + gx);
      asm volatile("global_load_async_to_lds_b32 %0, %1, off"
                   :: "v"(loff), "v"(gaddr) : "memory");
    } else {
      s_a[didx] = NEGS;                               // out-of-image pad
    }
  }
  asm volatile("s_wait_asynccnt 0x0" ::: "memory");
  __syncthreads();

  // ---- weight plane for min-pool of staged img: w = exp(-x/alpha) ---------
  for (int i = tid; i < STILE * STILE; i += 256) {
    const int ly = i / STILE, lx = i - ly * STILE;
    const int idx = ly * SP + lx;
    const float v = s_a[idx];
    s_w[idx] = (v > VALID) ? __expf(-INV_A * v) : 0.f;
  }
  __syncthreads();

  // ---- t1 over 68x68 (halo 2): erode (or identity for INIT) ---------------
  {
    const int R = STILE - 2;                          // 68
    for (int i = tid; i < R * R; i += 256) {
      const int ly = i / R + 1, lx = i % R + 1;
      const int idx = ly * SP + lx;
      const float c = s_a[idx];
      float t1;
      if (INIT)  t1 = c;                              // first open() on raw img
      else       t1 = (c > VALID) ? wavg9(s_a, s_w, idx) : NEGS;
      s_b[idx] = t1;
    }
  }
  __syncthreads();

  // ---- weights for min-pool of t1 -----------------------------------------
  {
    const int R = STILE - 2;
    for (int i = tid; i < R * R; i += 256) {
      const int ly = i / R + 1, lx = i % R + 1;
      const int idx = ly * SP + lx;
      const float v = s_b[idx];
      s_w[idx] = (v > VALID) ? __expf(-INV_A * v) : 0.f;
    }
  }
  __syncthreads();

  // ---- t2 = minpool(t1) over 66x66 (halo 1), into s_a ---------------------
  {
    const int R = STILE - 4;                          // 66
    for (int i = tid; i < R * R; i += 256) {
      const int ly = i / R + 2, lx = i % R + 2;
      const int idx = ly * SP + lx;
      const float c = s_b[idx];
      s_a[idx] = (c > VALID) ? wavg9(s_b, s_w, idx) : NEGS;
    }
  }
  __syncthreads();

  // ---- weights for MAX-pool of t2: w = exp(+x/alpha) ----------------------
  {
    const int R = STILE - 4;
    for (int i = tid; i < R * R; i += 256) {
      const int ly = i / R + 2, lx = i % R + 2;
      const int idx = ly * SP + lx;
      const float v = s_a[idx];
      s_w[idx] = (v > VALID) ? __expf(INV_A * v) : 0.f;
    }
  }
  __syncthreads();

  // ---- t3 = maxpool(t2) on 64x64 center; finalize skel update -------------
  for (int i = tid; i < TILE * TILE; i += 256) {
    const int ly = i / TILE + HALO, lx = i % TILE + HALO;
    const int idx = ly * SP + lx;
    const float t3 = wavg9(s_a, s_w, idx);            // open(img)
    const float t1 = s_b[idx];                        // eroded img (new img)
    const float delta = fmaxf(t1 - t3, 0.f);
    const int gy = y0 + ly, gx = x0 + lx;
    const size_t g = base + (size_t)gy * IMG_W + gx;
    float sk;
    if (INIT) {
      sk = delta;                                     // skel = relu(img-open)
    } else {
      const float s0 = skel_in[g];
      sk = s0 + fmaxf(delta - s0 * delta, 0.f);
    }
    img_out[g]  = t1;
    skel_out[g] = sk;
  }
}

// ---- deterministic two-pass reduction: ||skel_p - skel_t||_2 --------------
__global__ __launch_bounds__(256) void sqdiff_partial(
    const float* __restrict__ a, const float* __restrict__ c,
    float* __restrict__ partial, int n) {
  __shared__ float red[256];
  float acc = 0.f;
  for (int i = blockIdx.x * 256 + threadIdx.x; i < n; i += gridDim.x * 256) {
    const float d = a[i] - c[i];
    acc = fmaf(d, d, acc);
  }
  red[threadIdx.x] = acc;
  __syncthreads();
  for (int s = 128; s > 0; s >>= 1) {
    if (threadIdx.x < s) red[threadIdx.x] += red[threadIdx.x + s];
    __syncthreads();
  }
  if (threadIdx.x == 0) partial[blockIdx.x] = red[0];
}

__global__ __launch_bounds__(1024) void finalize_norm(
    const float* __restrict__ partial, int n, float* __restrict__ out) {
  __shared__ float red[1024];
  float acc = 0.f;
  for (int i = threadIdx.x; i < n; i += 1024) acc += partial[i];
  red[threadIdx.x] = acc;
  __syncthreads();
  for (int s = 512; s > 0; s >>= 1) {
    if (threadIdx.x < s) red[threadIdx.x] += red[threadIdx.x + s];
    __syncthreads();
  }
  if (threadIdx.x == 0) out[0] = sqrtf(red[0]);
}

extern "C" void kernel_launch(void* const* d_in, const int* in_sizes, int n_in,
                              void* d_out, int out_size, void* d_ws, size_t ws_size,
                              hipStream_t stream) {
  (void)in_sizes; (void)n_in; (void)out_size; (void)ws_size;
  constexpr size_t N = (size_t)IMG_B * IMG_W * IMG_H;   // 8Mi elements / image

  float* ws       = (float*)d_ws;
  float* imgA     = ws + 0 * N;
  float* imgB     = ws + 1 * N;
  float* skelA    = ws + 2 * N;
  float* skelB    = ws + 3 * N;
  float* skelC    = ws + 4 * N;
  float* skelD    = ws + 5 * N;
  float* partials = ws + 6 * N;                         // 2048 floats

  const dim3 grid((IMG_W / TILE) * (IMG_H / TILE) * IMG_B);  // 2048 tiles
  const dim3 block(256);

  auto run_image = [&](const float* src, float* skX, float* skY) {
    // init: skel0 = relu(img - open(img)); img copied into ping buffer A
    skel_iter<true><<<grid, block, 0, stream>>>(src, nullptr, imgA, skX);
    float* ia = imgA; float* ib = imgB;
    float* sa = skX;  float* sb = skY;
    for (int k = 0; k < 50; ++k) {
      skel_iter<false><<<grid, block, 0, stream>>>(ia, sa, ib, sb);
      float* t;
      t = ia; ia = ib; ib = t;
      t = sa; sa = sb; sb = t;
    }
    // 50 swaps (even) => final skeleton resides in skX
  };

  run_image((const float*)d_in[0], skelA, skelB);   // y_pred
  run_image((const float*)d_in[1], skelC, skelD);   // y_true

  sqdiff_partial<<<dim3(2048), dim3(256), 0, stream>>>(skelA, skelC, partials,
                                                       (int)N);
  finalize_norm<<<dim3(1), dim3(1024), 0, stream>>>(partials, 2048,
                                                    (float*)d_out);
}